// adaptive_aggregationPC_8443905704090
// MI455X (gfx1250) — compile-verified
//
#include <hip/hip_runtime.h>
#include <cstdint>

// ---------------------------------------------------------------------------
// adaptive aggregation: out[b,0,h,w] = sum_{dh,dw} max(thick,thin)[b,0,h+dh-2,w+dw-2]
//                                       * agg[b, dh*5+dw, h, w]      (zero pad)
// B=8, H=W=512, K=5.  Pure HBM-bandwidth-bound (~235 MB @ 0.45 FLOP/B -> ~10us
// floor at 23.3 TB/s).  Strategy: stream agg_coeff once with NT b128 loads,
// stage the reused sal halo tile in LDS via the CDNA5 async global->LDS path.
// ---------------------------------------------------------------------------

typedef float v4f __attribute__((ext_vector_type(4)));
typedef int   v2i __attribute__((ext_vector_type(2)));

constexpr int Hdim = 512;
constexpr int Wdim = 512;
constexpr int CCH  = 25;
constexpr int TILE_W = 64;
constexpr int TILE_H = 16;
constexpr int SW = TILE_W + 4;   // 68 (multiple of 4 -> 16B-aligned rows)
constexpr int SH = TILE_H + 4;   // 20
constexpr int NTHREADS = 256;    // 8 wave32s

// ---- CDNA5 async global->LDS copy (ASYNCcnt path) -------------------------
// Probe round 1 confirmed the builtin exists with param0 =
// "__attribute__((__vector_size__(2*sizeof(int)))) int __device__ *", i.e.
// v2i in the global address space; param1 is the LDS-side v2i pointer.
typedef __attribute__((address_space(1))) v2i* as1_v2i_ptr;
typedef __attribute__((address_space(3))) v2i* as3_v2i_ptr;

__device__ __forceinline__ void async_copy_b64_to_lds(void* lds, const void* g) {
#if __has_builtin(__builtin_amdgcn_global_load_async_to_lds_b64)
  __builtin_amdgcn_global_load_async_to_lds_b64(
      (as1_v2i_ptr)(uintptr_t)g,                 // global src (AS1), v2i pointee
      (as3_v2i_ptr)(uint32_t)(uintptr_t)lds,     // LDS dst (AS3): low 32 bits
      0, 0);                                     // imm offset, imm cpol
#else
  uint32_t l = (uint32_t)(uintptr_t)lds;
  asm volatile("global_load_async_to_lds_b64 %0, %1, off"
               :: "v"(l), "v"(g) : "memory");
#endif
}

__device__ __forceinline__ void async_wait_all() {
#if __has_builtin(__builtin_amdgcn_s_wait_asynccnt)
  __builtin_amdgcn_s_wait_asynccnt(0);
#else
  asm volatile("s_wait_asynccnt 0" ::: "memory");
#endif
}

// ---------------------------------------------------------------------------
__global__ __launch_bounds__(NTHREADS)
void adaptive_agg_kernel(const float* __restrict__ thick,
                         const float* __restrict__ thin,
                         const float* __restrict__ agg,
                         float* __restrict__ out)
{
  __shared__ __align__(16) float sThick[SH * SW];
  __shared__ __align__(16) float sThin [SH * SW];

  const int w0  = blockIdx.x * TILE_W;
  const int h0  = blockIdx.y * TILE_H;
  const int b   = blockIdx.z;
  const int tid = threadIdx.x;

  const size_t HW      = (size_t)Hdim * Wdim;
  const size_t planeOf = (size_t)b * HW;

  // Border tiles have halo cells outside the image: pre-zero LDS so skipped
  // async transfers leave correct zero padding.  (ds stores complete before
  // the barrier; async ops are issued after it -> ordering is safe.)
  const bool border = (h0 == 0) | (h0 + TILE_H == Hdim) |
                      (w0 == 0) | (w0 + TILE_W == Wdim);
  if (border) {
    for (int e = tid; e < SH * SW; e += NTHREADS) {
      sThick[e] = 0.0f;
      sThin [e] = 0.0f;
    }
    __syncthreads();
  }

  // ---- Stage halo tiles of thick & thin via async global->LDS b64 --------
  // Tile origin is (h0-2, w0-2); copy in float2 pairs: gcol = w0-2+2*p2 is
  // even -> 8B-aligned.  In-range test is uniform per pair (no partial pairs).
  for (int e = tid; e < SH * (SW / 2); e += NTHREADS) {
    const int row  = e / (SW / 2);
    const int p2   = e % (SW / 2);
    const int grow = h0 - 2 + row;
    const int gcol = w0 - 2 + 2 * p2;
    if ((unsigned)grow < (unsigned)Hdim && (unsigned)gcol < (unsigned)(Wdim - 1)) {
      const size_t g = planeOf + (size_t)grow * Wdim + (size_t)gcol;
      const int    l = row * SW + 2 * p2;
      async_copy_b64_to_lds(&sThick[l], thick + g);
      async_copy_b64_to_lds(&sThin [l], thin  + g);
    }
  }
  async_wait_all();
  __syncthreads();

  // ---- sal = max(thick, thin), in place in LDS ----------------------------
  for (int e = tid; e < SH * SW; e += NTHREADS)
    sThick[e] = fmaxf(sThick[e], sThin[e]);
  __syncthreads();

  // ---- Compute one float4 of output per thread ----------------------------
  const int r  = tid >> 4;          // 0..15 : row within tile
  const int c4 = (tid & 15) << 2;   // 0..60 : first col within tile
  const int h  = h0 + r;
  const int wb = w0 + c4;

  const float* aggB = agg + (size_t)b * CCH * HW + (size_t)h * Wdim + wb;
  __builtin_prefetch(aggB, 0, 3);   // gfx1250 global_prefetch_b8 on the agg stream

  float acc0 = 0.f, acc1 = 0.f, acc2 = 0.f, acc3 = 0.f;

#pragma unroll
  for (int dh = 0; dh < 5; ++dh) {
    // Aligned 16B LDS reads: row offset (r+dh)*68 and col c4 are multiples of 4.
    const float* srow = &sThick[(r + dh) * SW + c4];
    const v4f s0 = *(const v4f*)(srow);
    const v4f s1 = *(const v4f*)(srow + 4);
    float s[8];
#pragma unroll
    for (int j = 0; j < 4; ++j) { s[j] = s0[j]; s[4 + j] = s1[j]; }

#pragma unroll
    for (int dw = 0; dw < 5; ++dw) {
      // agg_coeff: 210 MB single-use stream -> non-temporal b128 loads.
      const v4f a = __builtin_nontemporal_load(
          (const v4f*)(aggB + (size_t)(dh * 5 + dw) * HW));
      acc0 = fmaf(a[0], s[dw + 0], acc0);
      acc1 = fmaf(a[1], s[dw + 1], acc1);
      acc2 = fmaf(a[2], s[dw + 2], acc2);
      acc3 = fmaf(a[3], s[dw + 3], acc3);
    }
  }

  const v4f o = {acc0, acc1, acc2, acc3};
  __builtin_nontemporal_store(o, (v4f*)(out + planeOf + (size_t)h * Wdim + wb));
}

// ---------------------------------------------------------------------------
extern "C" void kernel_launch(void* const* d_in, const int* in_sizes, int n_in,
                              void* d_out, int out_size, void* d_ws, size_t ws_size,
                              hipStream_t stream) {
  (void)n_in; (void)out_size; (void)d_ws; (void)ws_size;
  const float* thick = (const float*)d_in[0];
  const float* thin  = (const float*)d_in[1];
  const float* agg   = (const float*)d_in[2];
  float*       out   = (float*)d_out;

  const int B = in_sizes[2] / (CCH * Hdim * Wdim);   // = 8

  dim3 grid(Wdim / TILE_W, Hdim / TILE_H, B);        // (8, 32, 8)
  dim3 block(NTHREADS);
  adaptive_agg_kernel<<<grid, block, 0, stream>>>(thick, thin, agg, out);
}